// mHC_8718783611220
// MI455X (gfx1250) — compile-verified
//
#include <hip/hip_runtime.h>
#include <hip/hip_bf16.h>

typedef float v2f __attribute__((ext_vector_type(2)));
typedef float v8f __attribute__((ext_vector_type(8)));

#define NB      4
#define DIM     2048
#define NC      (NB * DIM)       // 8192
#define NOUT    24               // n^2 + 2n
#define TILE_M  16               // tokens per workgroup (= WMMA M)
#define BLOCK   256
#define WAVES   (BLOCK / 32)     // 8
#define KSLICE  (NC / WAVES)     // 1024 contiguous K per wave
#define NQ      (DIM / 4)        // float4 groups per branch row

__global__ __launch_bounds__(BLOCK) void mhc_fused_kernel(
    const float* __restrict__ hs,    // [T, NB, DIM] == [T, NC]
    const float* __restrict__ Wphi,  // [NOUT, NC]
    const float* __restrict__ avec,  // [3]
    const float* __restrict__ bvec,  // [NOUT]
    float* __restrict__ out)         // [T, NB, DIM]
{
    __shared__ float ldsHp[WAVES][TILE_M][32]; // per-wave partial H (16 KB)
    __shared__ float ldsSq[WAVES][32];         // per-wave partial sum-of-squares
    __shared__ float Hfin[TILE_M][NOUT];
    __shared__ float rinv_s[TILE_M];
    __shared__ float cPre[TILE_M][NB];
    __shared__ float cPost[TILE_M][NB];
    __shared__ float cRes[TILE_M][NB * NB];

    const int tid  = threadIdx.x;
    const int wave = tid >> 5;
    const int lane = tid & 31;
    const int g    = lane >> 4;   // half-wave group (selects K sub-pair)
    const int m    = lane & 15;   // token row (A) / output col (B) within tile
    const long tile0 = (long)blockIdx.x * TILE_M;

    // ---------------- Phase 1: H[16,24] = hs_tile[16,NC] x Wphi^T via WMMA f32 16x16x4
    const float* aptr  = hs   + (tile0 + m) * (long)NC + wave * KSLICE + 2 * g;
    const float* b0ptr = Wphi + (long)m * NC           + wave * KSLICE + 2 * g;
    const int  col1    = 16 + m;
    const int  col1c   = (col1 < NOUT) ? col1 : (NOUT - 1);
    const float b1mask = (col1 < NOUT) ? 1.0f : 0.0f;
    const float* b1ptr = Wphi + (long)col1c * NC       + wave * KSLICE + 2 * g;

    v8f acc0 = {0.f, 0.f, 0.f, 0.f, 0.f, 0.f, 0.f, 0.f};
    v8f acc1 = {0.f, 0.f, 0.f, 0.f, 0.f, 0.f, 0.f, 0.f};
    float sq = 0.0f;

    for (int k = 0; k < KSLICE; k += 4) {
        v2f av = *(const v2f*)(aptr + k);   // A: token m, K = k + 2g + {0,1}
        v2f b0 = *(const v2f*)(b0ptr + k);  // B: out col m
        v2f b1 = *(const v2f*)(b1ptr + k);  // B: out col 16+m (masked past 23)
        b1.x *= b1mask;
        b1.y *= b1mask;
        acc0 = __builtin_amdgcn_wmma_f32_16x16x4_f32(false, av, false, b0,
                                                     (short)0, acc0, false, false);
        acc1 = __builtin_amdgcn_wmma_f32_16x16x4_f32(false, av, false, b1,
                                                     (short)0, acc1, false, false);
        sq += av.x * av.x + av.y * av.y;    // norm accumulation (each k hit once)
    }

    // D layout: lane l, vgpr i -> M = i + 8*(l/16), N = l%16
#pragma unroll
    for (int i = 0; i < 8; ++i) {
        ldsHp[wave][i + 8 * g][m]      = acc0[i];
        ldsHp[wave][i + 8 * g][16 + m] = acc1[i];
    }
    ldsSq[wave][lane] = sq;
    __syncthreads();

    // Cross-wave reduction of H (16x24 entries)
    for (int e = tid; e < TILE_M * NOUT; e += BLOCK) {
        const int row = e / NOUT, co = e % NOUT;
        float s = 0.0f;
#pragma unroll
        for (int w = 0; w < WAVES; ++w) s += ldsHp[w][row][co];
        Hfin[row][co] = s;
    }
    if (tid < TILE_M) {
        float s = 0.0f;
#pragma unroll
        for (int w = 0; w < WAVES; ++w) s += ldsSq[w][tid] + ldsSq[w][tid + 16];
        rinv_s[tid] = sqrtf((float)NC / s);   // 1/r = sqrt(nc)/||x||
    }
    __syncthreads();

    // ---------------- Per-token activations + Sinkhorn (one lane per token)
    if (tid < TILE_M) {
        const float rinv = rinv_s[tid];
        const float a0 = avec[0], a1 = avec[1], a2 = avec[2];
#pragma unroll
        for (int i = 0; i < NB; ++i) {
            cPre[tid][i]  = 1.0f / (1.0f + expf(-(rinv * Hfin[tid][i] * a0 + bvec[i])));
            cPost[tid][i] = 2.0f / (1.0f + expf(-(rinv * Hfin[tid][NB + i] * a1 + bvec[NB + i])));
        }
        float K[16];
#pragma unroll
        for (int i = 0; i < 16; ++i)
            K[i] = expf(rinv * Hfin[tid][2 * NB + i] * a2 + bvec[2 * NB + i]);
#pragma unroll 1
        for (int it = 0; it < 20; ++it) {
#pragma unroll
            for (int r = 0; r < 4; ++r) {
                const float is = 1.0f / (K[r*4] + K[r*4+1] + K[r*4+2] + K[r*4+3] + 1e-20f);
                K[r*4] *= is; K[r*4+1] *= is; K[r*4+2] *= is; K[r*4+3] *= is;
            }
#pragma unroll
            for (int c = 0; c < 4; ++c) {
                const float is = 1.0f / (K[c] + K[4+c] + K[8+c] + K[12+c] + 1e-20f);
                K[c] *= is; K[4+c] *= is; K[8+c] *= is; K[12+c] *= is;
            }
        }
#pragma unroll
        for (int i = 0; i < 16; ++i) cRes[tid][i] = K[i];
    }
    __syncthreads();

    // ---------------- Phase 2: recombination (hs re-read is L2-hot), coalesced float4
    for (int f = tid; f < TILE_M * NQ; f += BLOCK) {
        const int tk = f / NQ;
        const int q  = f % NQ;
        const float4* xb = (const float4*)(hs + (tile0 + tk) * (long)NC) + q;
        const float4 x0 = xb[0 * NQ];
        const float4 x1 = xb[1 * NQ];
        const float4 x2 = xb[2 * NQ];
        const float4 x3 = xb[3 * NQ];

        const float p0 = cPre[tk][0], p1 = cPre[tk][1], p2 = cPre[tk][2], p3 = cPre[tk][3];
        float4 hp;
        hp.x = p0 * x0.x + p1 * x1.x + p2 * x2.x + p3 * x3.x;
        hp.y = p0 * x0.y + p1 * x1.y + p2 * x2.y + p3 * x3.y;
        hp.z = p0 * x0.z + p1 * x1.z + p2 * x2.z + p3 * x3.z;
        hp.w = p0 * x0.w + p1 * x1.w + p2 * x2.w + p3 * x3.w;

        float4* ob = (float4*)(out + (tile0 + tk) * (long)NC) + q;
#pragma unroll
        for (int mr = 0; mr < NB; ++mr) {
            const float r0 = cRes[tk][mr * 4 + 0];
            const float r1 = cRes[tk][mr * 4 + 1];
            const float r2 = cRes[tk][mr * 4 + 2];
            const float r3 = cRes[tk][mr * 4 + 3];
            const float po = cPost[tk][mr];
            float4 o;
            o.x = po * hp.x + r0 * x0.x + r1 * x1.x + r2 * x2.x + r3 * x3.x;
            o.y = po * hp.y + r0 * x0.y + r1 * x1.y + r2 * x2.y + r3 * x3.y;
            o.z = po * hp.z + r0 * x0.z + r1 * x1.z + r2 * x2.z + r3 * x3.z;
            o.w = po * hp.w + r0 * x0.w + r1 * x1.w + r2 * x2.w + r3 * x3.w;
            ob[mr * NQ] = o;
        }
    }
}

extern "C" void kernel_launch(void* const* d_in, const int* in_sizes, int n_in,
                              void* d_out, int out_size, void* d_ws, size_t ws_size,
                              hipStream_t stream) {
    const float* hs   = (const float*)d_in[0];  // [B, L, NB, DIM] fp32
    const float* Wphi = (const float*)d_in[1];  // [NOUT, NC] fp32
    const float* a    = (const float*)d_in[2];  // [3]
    const float* b    = (const float*)d_in[3];  // [NOUT]
    float* out        = (float*)d_out;

    const int T = in_sizes[0] / NC;             // 8192 tokens
    const int grid = T / TILE_M;                // 512 workgroups
    mhc_fused_kernel<<<grid, BLOCK, 0, stream>>>(hs, Wphi, a, b, out);
}